// RecurrentNeuralNetwork_4355096838397
// MI455X (gfx1250) — compile-verified
//
#include <hip/hip_runtime.h>
#include <math.h>
#include <stdint.h>

// ---------------------------------------------------------------------------
// RNN with per-batch plastic recurrent weights (B=16, T=64, H=256).
// 1 workgroup per batch; diff_j (= new_j - w_hh, 256KB) LDS-resident.
// Per step:
//   - syn_noise[t,b] (256KB) streamed HBM -> LDS with double-buffered
//     GLOBAL_LOAD_ASYNC_TO_LDS_B128 (ASYNCcnt), overlapped with compute
//   - matvecs (act@w_hh.T, act . diff_j) via VALU against LDS
//   - plasticity  diff += beta*(-act (x) act + sn*sigma*sqrt(beta))
//     via V_WMMA_F32_16X16X4_F32 rank-1 outer-product tiles
// ---------------------------------------------------------------------------

typedef float v2f __attribute__((ext_vector_type(2)));
typedef float v8f __attribute__((ext_vector_type(8)));

#define BN   16
#define TN   64
#define HN   256
#define NIN  2
#define NOUT 2

#define ALPHA      0.25f
#define BETA       0.1f
#define SIGMA_SYN  0.002f
#define PERT_SIGMA 0.1f
#define SQRT_ALPHA 0.5f                  // sqrt(0.25)
#define SQRT_BETA  0.31622776601683794f  // sqrt(0.1)

// LDS layout (dynamic shared, assumed base offset 0):
//   [0, HN*HN)            diff_j
//   [HN*HN, +HN)          act
//   [HN*HN+HN, +HN)       h
//   [HN*HN+2HN, +4096)    per-wave sn staging: 8 waves x 2 bufs x 256 floats
#define STAGE_ELEMS (HN * HN + 2 * HN)
#define STAGE_BYTES ((uint32_t)(STAGE_ELEMS * 4))
#define LDS_FLOATS  (STAGE_ELEMS + 8 * 2 * 256)

// Issue async copy of one 16x16 f32 tile (1KB) of sn into this wave's staging
// buffer (buf = tt&1). 2 x b128: 32 lanes x 16B = 512B each (rows 0-7, 8-15).
__device__ __forceinline__ void stage_tile(const float* __restrict__ snb,
                                           int wave, int lane, int tt) {
    const int i0  = (((wave << 1) + (tt >> 4)) << 4);
    const int j0  = ((tt & 15) << 4);
    const int row = lane >> 2;          // 0..7
    const int c4  = (lane & 3) << 2;    // 0,4,8,12
    const float* g0 = snb + (size_t)(i0 + row) * HN + j0 + c4;
    const float* g1 = g0 + 8 * HN;
    const uint32_t l0 = STAGE_BYTES + ((uint32_t)wave << 11) +
                        ((uint32_t)(tt & 1) << 10) + ((uint32_t)((row << 4) + c4) << 2);
    const uint32_t l1 = l0 + 512u;
    asm volatile("global_load_async_to_lds_b128 %0, %1, off"
                 :: "v"(l0), "v"((uint64_t)(uintptr_t)g0) : "memory");
    asm volatile("global_load_async_to_lds_b128 %0, %1, off"
                 :: "v"(l1), "v"((uint64_t)(uintptr_t)g1) : "memory");
}

__global__ __launch_bounds__(256, 1)
void rnn_plastic_kernel(const float* __restrict__ x,          // [B,T,NIN]
                        const float* __restrict__ h0,         // [B,H]
                        const float* __restrict__ w_in,       // [H,NIN]
                        const float* __restrict__ w_hh,       // [H,H]
                        const float* __restrict__ b_hh,       // [H]
                        const float* __restrict__ w_out,      // [NOUT,H]
                        const float* __restrict__ noise_pert, // [B,H]
                        const float* __restrict__ syn,        // [T,B,H,H]
                        const int*   __restrict__ ptime,      // scalar
                        float* __restrict__ out)              // concat outputs
{
    extern __shared__ float smem[];
    float* diff  = smem;                   // [H*H]
    float* act   = smem + HN * HN;         // [H]
    float* hld   = act + HN;               // [H]
    float* stage = hld + HN;               // [8*2*256]

    const int b    = blockIdx.x;
    const int tid  = threadIdx.x;          // 0..255, tid == hidden index j
    const int lane = tid & 31;
    const int wave = tid >> 5;             // 0..7
    const int half = lane >> 4;            // 0 or 1
    const int l16  = lane & 15;
    const int pt   = *ptime;

    float* stage_w = stage + (wave << 9);  // this wave's 2 x 256-float buffers

    // ---- output layout (flat concat, reference return order) ----
    float* hid_out  = out;                                 // [B,T,H]
    float* outl_out = out + (size_t)BN * TN * HN;          // [B,T,NOUT]
    float* hfin_out = outl_out + (size_t)BN * TN * NOUT;   // [B,H]
    float* nj_out   = hfin_out + (size_t)BN * HN;          // [B,H,H]

    // ---- init: diff_j = 0 (new_j starts at w_hh), h = h0 ----
    for (int k = tid; k < HN * HN; k += 256) diff[k] = 0.0f;
    hld[tid] = h0[b * HN + tid];

    // per-thread constants for row j = tid
    const float bhh_j = b_hh[tid];
    const float wi0   = w_in[tid * NIN + 0];
    const float wi1   = w_in[tid * NIN + 1];
    const float np_j  = noise_pert[b * HN + tid] * (PERT_SIGMA * SQRT_ALPHA);
    const float snsc  = SIGMA_SYN * SQRT_BETA;
    __syncthreads();

    for (int t = 0; t < TN; ++t) {
        const float* __restrict__ snb = syn + ((size_t)t * BN + b) * (HN * HN);

        // kick off async staging of this wave's first two tiles; the DMA
        // overlaps the activation + matvec phase below
        stage_tile(snb, wave, lane, 0);
        stage_tile(snb, wave, lane, 1);

        // prefetch next step's noise toward L2
        if (t + 1 < TN) {
            const float* snn = syn + ((size_t)(t + 1) * BN + b) * (HN * HN);
            __builtin_prefetch(snn + (size_t)tid * 128, 0, 1);
            __builtin_prefetch(snn + (size_t)tid * 128 + 64, 0, 1);
        }

        // ---- activation ----
        const float hj = hld[tid];
        const float aj = tanhf(hj);
        act[tid] = aj;
        __syncthreads();

        // ---- matvecs: hh[j] = sum_i act[i]*w_hh[j,i]   (act @ w_hh.T)
        // ----          dj[j] = sum_i act[i]*diff[i,j]   (pre-update diff_j)
        float dj = 0.0f, hh = 0.0f;
        const float4* __restrict__ wrow = (const float4*)(w_hh + (size_t)tid * HN);
        #pragma unroll 4
        for (int i = 0; i < HN; i += 4) {
            const float4 w4 = wrow[i >> 2];
            const float a0 = act[i + 0], a1 = act[i + 1];
            const float a2 = act[i + 2], a3 = act[i + 3];
            hh = fmaf(w4.x, a0, hh); hh = fmaf(w4.y, a1, hh);
            hh = fmaf(w4.z, a2, hh); hh = fmaf(w4.w, a3, hh);
            dj = fmaf(a0, diff[(i + 0) * HN + tid], dj);
            dj = fmaf(a1, diff[(i + 1) * HN + tid], dj);
            dj = fmaf(a2, diff[(i + 2) * HN + tid], dj);
            dj = fmaf(a3, diff[(i + 3) * HN + tid], dj);
        }

        // ---- input projection + leaky update ----
        const float x0 = x[((size_t)b * TN + t) * NIN + 0];
        const float x1 = x[((size_t)b * TN + t) * NIN + 1];
        const float tmp   = fmaf(x0, wi0, fmaf(x1, wi1, hh + bhh_j + dj));
        const float noise = (t == pt) ? np_j : 0.0f;
        const float hn    = fmaf(1.0f - ALPHA, hj, fmaf(ALPHA, tmp, noise));

        __syncthreads();  // all dj reads of diff complete before update
        hld[tid] = hn;
        hid_out[((size_t)b * TN + t) * HN + tid] = hn;

        // ---- plasticity: diff += beta*(-act(x)act + sn*sigma*sqrt(beta))
        // 16x16 outer tiles via V_WMMA_F32_16X16X4_F32 (K row/col 0 only);
        // sn consumed from the double-buffered async LDS staging.
        for (int tt = 0; tt < 32; ++tt) {
            // wait for tile tt (older of the pending tiles; loads are in-order)
            if (tt + 1 < 32)
                asm volatile("s_wait_asynccnt 0x2" ::: "memory");
            else
                asm volatile("s_wait_asynccnt 0x0" ::: "memory");

            const int i0 = (((wave << 1) + (tt >> 4)) << 4);
            const int j0 = ((tt & 15) << 4);
            const float ar = act[i0 + l16];   // A rows (M = lane 0..15)
            const float ac = act[j0 + l16];   // B cols (N = lane 0..15)
            v2f A  = {half ? 0.0f : ar, 0.0f};
            v2f Bm = {half ? 0.0f : ac, 0.0f};
            v8f C  = {};
            v8f D  = __builtin_amdgcn_wmma_f32_16x16x4_f32(
                         false, A, false, Bm, (short)0, C, false, false);
            // D lane layout: VGPR v, lane l -> row i0+v+8*(l>=16), col j0+l%16
            const float* sb   = stage_w + ((tt & 1) << 8);
            const int    col  = j0 + l16;
            const int    rb   = (half << 3);
            #pragma unroll
            for (int v = 0; v < 8; ++v) {
                const float sn_v = sb[((rb + v) << 4) + l16];
                const int   idx  = (i0 + rb + v) * HN + col;
                diff[idx] += BETA * fmaf(sn_v, snsc, -D[v]);
            }
            // refill the buffer just consumed with tile tt+2
            if (tt + 2 < 32) stage_tile(snb, wave, lane, tt + 2);
        }
        __syncthreads();

        // ---- readout: out = h_new @ w_out.T (waves 0,1; shfl reduction) ----
        if (wave < NOUT) {
            const float* __restrict__ wo = w_out + (size_t)wave * HN;
            float acc = 0.0f;
            #pragma unroll
            for (int jj = lane; jj < HN; jj += 32)
                acc = fmaf(wo[jj], hld[jj], acc);
            #pragma unroll
            for (int m = 16; m > 0; m >>= 1)
                acc += __shfl_xor(acc, m, 32);
            if (lane == 0)
                outl_out[((size_t)b * TN + t) * NOUT + wave] = acc;
        }
        __syncthreads();
    }

    // ---- final state outputs ----
    hfin_out[b * HN + tid] = hld[tid];
    const size_t njb = (size_t)b * HN * HN;
    for (int k = tid; k < HN * HN; k += 256)
        nj_out[njb + k] = diff[k] + w_hh[k];   // new_j = w_hh + diff
}

extern "C" void kernel_launch(void* const* d_in, const int* in_sizes, int n_in,
                              void* d_out, int out_size, void* d_ws, size_t ws_size,
                              hipStream_t stream) {
    (void)in_sizes; (void)n_in; (void)out_size; (void)d_ws; (void)ws_size;

    const float* x     = (const float*)d_in[0];
    const float* h0    = (const float*)d_in[1];
    const float* w_in  = (const float*)d_in[2];
    const float* w_hh  = (const float*)d_in[3];
    const float* b_hh  = (const float*)d_in[4];
    const float* w_out = (const float*)d_in[5];
    const float* npert = (const float*)d_in[6];
    const float* syn   = (const float*)d_in[7];
    const int*   pt    = (const int*)d_in[8];
    float*       out   = (float*)d_out;

    const size_t shmem = (size_t)LDS_FLOATS * sizeof(float); // ~280.5 KB of 320 KB
    (void)hipFuncSetAttribute(reinterpret_cast<const void*>(&rnn_plastic_kernel),
                              hipFuncAttributeMaxDynamicSharedMemorySize,
                              (int)shmem);

    rnn_plastic_kernel<<<BN, 256, shmem, stream>>>(
        x, h0, w_in, w_hh, b_hh, w_out, npert, syn, pt, out);
}